// MeanSquaredError2_51178830299666
// MI455X (gfx1250) — compile-verified
//
#include <hip/hip_runtime.h>

// ---------------------------------------------------------------------------
// MeanSquaredError2: out = ( sum((h-tt)^2) + sum((o-pp)^2) ) / (sum(v)/2)
//   o: [B, 28, 14, 14] f32   h: [B, 14, 14, 14] f32
//   t: [B, 14, 2]      f32   v: [B, 14, 1]      f32
// Memory-bound: ~542 MB read -> ~23us at 23.3 TB/s. Targets tt/pp are
// regenerated on the fly from t/v (L2-resident), main stream uses b128
// non-temporal loads, wave32 shuffle + LDS block reduction, and an f32 WMMA
// accumulate loop (branch-free, b64 loads) for the cross-block reduction.
// ---------------------------------------------------------------------------

typedef __attribute__((ext_vector_type(4))) float v4f;
typedef __attribute__((ext_vector_type(2))) float v2f;
typedef __attribute__((ext_vector_type(8))) float v8f;

#define NBLK 4096       // multiple of 64 -> tail-free WMMA reduction
#define NTHREADS 256

__global__ __launch_bounds__(NTHREADS) void mse2_partials_kernel(
    const float* __restrict__ o, const float* __restrict__ h,
    const float* __restrict__ t, const float* __restrict__ v,
    unsigned Bn, float* __restrict__ pS, float* __restrict__ pV) {
  const unsigned gid    = blockIdx.x * NTHREADS + threadIdx.x;
  const unsigned stride = NBLK * NTHREADS;

  const unsigned No4 = Bn * 28u * 49u;  // o as float4  (slice = 49 float4)
  const unsigned Nh4 = Bn * 14u * 49u;  // h as float4
  const unsigned Nv  = Bn * 14u;        // visibility flags

  const v4f* __restrict__ o4 = (const v4f*)o;
  const v4f* __restrict__ h4 = (const v4f*)h;

  float acc = 0.0f;  // d1 + d2 partial
  float av  = 0.0f;  // sum(v) partial

  // ---- Phase 1: sum((o - pp)^2), streaming o with b128 NT loads ----
  for (unsigned q = gid; q < No4; q += stride) {
    unsigned s  = q / 49u;          // slice id = b*28 + j2
    unsigned r  = q - s * 49u;      // float4 within slice
    unsigned b  = s / 28u;
    unsigned j2 = s - b * 28u;
    bool     isy = j2 >= 14u;
    unsigned j   = isy ? (j2 - 14u) : j2;
    // tpos (pixel coords); t is tiny and stays L2/WGP$-resident
    float tc = t[(b * 14u + j) * 2u + (isy ? 1u : 0u)] * 224.0f;

    v4f ov = __builtin_nontemporal_load(o4 + q);
    unsigned p0 = 4u * r;
#pragma unroll
    for (int k = 0; k < 4; ++k) {
      unsigned p = p0 + (unsigned)k;  // p = a*14 + c, p in [0,196)
      unsigned a = p / 14u;           // magic-mul div by constant
      unsigned c = p - a * 14u;
      float idxf = (float)(isy ? a : c);
      float pp   = __builtin_fmaf(-14.0f, idxf, tc);  // tpos - 14*idx
      float d    = ov[k] - pp;
      acc = __builtin_fmaf(d, d, acc);
    }
  }

  // ---- Phase 2: sum((h - tt)^2), tt is one-hot at (xi, yi) if visible ----
  for (unsigned q = gid; q < Nh4; q += stride) {
    unsigned s = q / 49u;           // slice id = b*14 + j
    unsigned r = q - s * 49u;
    float txg = t[2u * s] * 14.0f;  // grid coords
    float tyg = t[2u * s + 1u] * 14.0f;
    int vis = ((int)v[s] == 1);
    int xi = (int)txg;  xi = xi < 0 ? 0 : (xi > 13 ? 13 : xi);
    int yi = (int)tyg;  yi = yi < 0 ? 0 : (yi > 13 ? 13 : yi);
    int code = vis ? (xi * 14 + yi) : -1;  // flat hot index, -1 if invisible

    v4f hv = __builtin_nontemporal_load(h4 + q);
    int p0 = 4 * (int)r;
#pragma unroll
    for (int k = 0; k < 4; ++k) {
      float tt = ((p0 + k) == code) ? 1.0f : 0.0f;
      float d  = hv[k] - tt;
      acc = __builtin_fmaf(d, d, acc);
    }
  }

  // ---- Phase 3: sum(v) ----
  for (unsigned i = gid; i < Nv; i += stride) av += v[i];

  // ---- wave32 butterfly reduce ----
#pragma unroll
  for (int m = 16; m; m >>= 1) {
    acc += __shfl_xor(acc, m, 32);
    av  += __shfl_xor(av,  m, 32);
  }

  // ---- cross-wave (8 waves) reduce via LDS, one partial pair per block ----
  __shared__ float sS[NTHREADS / 32];
  __shared__ float sV[NTHREADS / 32];
  unsigned wid = threadIdx.x >> 5, ln = threadIdx.x & 31u;
  if (ln == 0) { sS[wid] = acc; sV[wid] = av; }
  __syncthreads();
  if (threadIdx.x == 0) {
    float a = 0.0f, b2 = 0.0f;
#pragma unroll
    for (int w = 0; w < NTHREADS / 32; ++w) { a += sS[w]; b2 += sV[w]; }
    pS[blockIdx.x] = a;
    pV[blockIdx.x] = b2;
  }
}

// ---------------------------------------------------------------------------
// Final reduction: one wave32. The NBLK partials are reduced with an exact
// f32 WMMA accumulate loop: D = ones(16x4) * B(4x16) + D. With all-ones A,
// every D entry is a column sum of B, so Sum(all 256 D entries) equals
// 16 * Sum(all B values fed in) regardless of the B/D lane layouts — so each
// lane can simply grab two adjacent floats (one b64 load, branch-free since
// NBLK % 64 == 0).
// ---------------------------------------------------------------------------
__global__ __launch_bounds__(32) void mse2_finish_kernel(
    const float* __restrict__ pS, const float* __restrict__ pV,
    float* __restrict__ out) {
  const int lane = threadIdx.x;  // exactly one wave32, EXEC all ones

  v8f acc = {};
  v2f ones; ones.x = 1.0f; ones.y = 1.0f;

  const v2f* __restrict__ pS2 = (const v2f*)pS;  // NBLK/2 pairs
#pragma unroll 4
  for (int it = 0; it < NBLK / 64; ++it) {
    // 64 consecutive partials per iteration: lane -> pairs {2*lane, 2*lane+1}
    v2f bv = pS2[it * 32 + lane];  // single global_load_b64, no predication
    // v_wmma_f32_16x16x4_f32: D = A(ones) x B(partials) + C(acc)
    acc = __builtin_amdgcn_wmma_f32_16x16x4_f32(
        false, ones, false, bv, (short)0, acc, false, false);
  }

  float s = acc[0] + acc[1] + acc[2] + acc[3] +
            acc[4] + acc[5] + acc[6] + acc[7];
#pragma unroll
  for (int m = 16; m; m >>= 1) s += __shfl_xor(s, m, 32);
  s *= (1.0f / 16.0f);  // D holds 16 identical rows -> sum(D) = 16 * total

  // sum(v) partials: b64 per lane, branch-free, then butterfly
  const v2f* __restrict__ pV2 = (const v2f*)pV;
  v2f vp = {};
#pragma unroll 4
  for (int it = 0; it < NBLK / 64; ++it) {
    v2f x = pV2[it * 32 + lane];
    vp.x += x.x;
    vp.y += x.y;
  }
  float vsum = vp.x + vp.y;
#pragma unroll
  for (int m = 16; m; m >>= 1) vsum += __shfl_xor(vsum, m, 32);

  if (lane == 0) out[0] = 2.0f * s / vsum;  // (d1+d2) / (sum(v)/2)
}

extern "C" void kernel_launch(void* const* d_in, const int* in_sizes, int n_in,
                              void* d_out, int out_size, void* d_ws, size_t ws_size,
                              hipStream_t stream) {
  const float* o = (const float*)d_in[0];
  const float* h = (const float*)d_in[1];
  const float* t = (const float*)d_in[2];
  const float* v = (const float*)d_in[3];

  unsigned Nv = (unsigned)in_sizes[3];  // B * NJ
  unsigned Bn = Nv / 14u;

  float* pS = (float*)d_ws;        // NBLK partials of (d1+d2)
  float* pV = pS + NBLK;           // NBLK partials of sum(v)

  mse2_partials_kernel<<<NBLK, NTHREADS, 0, stream>>>(o, h, t, v, Bn, pS, pV);
  mse2_finish_kernel<<<1, 32, 0, stream>>>(pS, pV, (float*)d_out);
}